// BasePostProcess_44341242364825
// MI455X (gfx1250) — compile-verified
//
#include <hip/hip_runtime.h>
#include <hip/hip_bf16.h>

#define BATCH 4
#define A_    9
#define C_    80
#define KTOT  120087
#define SL1_BETA (1.0f / 9.0f)

typedef float v2f __attribute__((ext_vector_type(2)));
typedef float v8f __attribute__((ext_vector_type(8)));

// Exact f32 reduction of 32 lane values using the CDNA5 matrix pipe.
// A[m][k]: lane's value in one K slot, 0 in the other; B = all ones.
// D[m][*] = rowsum_m = v_m + v_{m+16}. Summing the 8 D components per lane
// gives S_low (lanes 0-15) / S_high (lanes 16-31); shfl_xor(16) completes it.
__device__ __forceinline__ float wave_sum32(float v) {
    v2f a;  a[0] = v;    a[1] = 0.0f;
    v2f bm; bm[0] = 1.0f; bm[1] = 1.0f;
    v8f c = {};
    v8f d = __builtin_amdgcn_wmma_f32_16x16x4_f32(
        /*neg_a=*/false, a, /*neg_b=*/false, bm,
        /*c_mod=*/(short)0, c, /*reuse_a=*/false, /*reuse_b=*/false);
    float s = ((d[0] + d[1]) + (d[2] + d[3])) + ((d[4] + d[5]) + (d[6] + d[7]));
    s += __shfl_xor(s, 16, 32);
    return s;
}

// Block (256 threads = 8 waves) reduction of 4 quantities -> one ws slot.
// Must be reached by ALL threads of the block (EXEC all ones for WMMA).
__device__ __forceinline__ void block_accum(float q0, float q1, float q2, float q3,
                                            float* __restrict__ slot) {
    __shared__ float lds[8 * 4];
    q0 = wave_sum32(q0);
    q1 = wave_sum32(q1);
    q2 = wave_sum32(q2);
    q3 = wave_sum32(q3);
    const int lane = threadIdx.x & 31;
    const int wv   = threadIdx.x >> 5;
    if (lane == 0) {
        lds[wv * 4 + 0] = q0;
        lds[wv * 4 + 1] = q1;
        lds[wv * 4 + 2] = q2;
        lds[wv * 4 + 3] = q3;
    }
    __syncthreads();
    if (threadIdx.x == 0) {
        float s0 = 0.f, s1 = 0.f, s2 = 0.f, s3 = 0.f;
        #pragma unroll
        for (int i = 0; i < 8; ++i) {   // fixed order -> deterministic
            s0 += lds[i * 4 + 0];
            s1 += lds[i * 4 + 1];
            s2 += lds[i * 4 + 2];
            s3 += lds[i * 4 + 3];
        }
        slot[0] = s0; slot[1] = s1; slot[2] = s2; slot[3] = s3;
    }
}

__device__ __forceinline__ float smooth_l1(float d) {
    d = fabsf(d);
    return (d < SL1_BETA) ? (0.5f * d * d / SL1_BETA) : (d - 0.5f * SL1_BETA);
}

__global__ __launch_bounds__(256)
void qfl_level_kernel(const float* __restrict__ cls_p,
                      const float* __restrict__ loc_p,
                      const float* __restrict__ anchors,
                      const float* __restrict__ loc_tgt,
                      const int*   __restrict__ cls_tgt,
                      const unsigned char* __restrict__ loc_mask,
                      float* __restrict__ slots, int slot_base,
                      int h, int w, int k_base) {
    const int tid = blockIdx.x * 256 + threadIdx.x;
    const int hw  = h * w;
    const int n   = BATCH * A_ * hw;

    float clsL = 0.f, locL = 0.f, accC = 0.f, posC = 0.f;

    if (tid < n) {
        // thread layout [b][a][y][x], x fastest -> coalesced over x
        int x = tid % w;
        int r = tid / w;
        int y = r % h;  r /= h;
        int a = r % A_;
        int b = r / A_;

        const int k  = k_base + (y * w + x) * A_ + a;
        const int bk = b * KTOT + k;

        const float m = loc_mask[bk] ? 1.0f : 0.0f;
        const int label = cls_tgt[bk];

        const size_t sp     = (size_t)y * w + x;
        const size_t stride = (size_t)hw;

        // ---- loc branch: gather 4 deltas (coalesced per component) ----
        const float* locptr = loc_p + (size_t)(b * A_ * 4 + a * 4) * stride + sp;
        const float d0 = locptr[0];
        const float d1 = locptr[stride];
        const float d2 = locptr[2 * stride];
        const float d3 = locptr[3 * stride];

        const float4 anc = ((const float4*)anchors)[k];
        const float4 tg  = ((const float4*)loc_tgt)[bk];

        const float aw  = anc.z - anc.x;
        const float ah  = anc.w - anc.y;
        const float acx = anc.x + 0.5f * aw;
        const float acy = anc.y + 0.5f * ah;

        // decode predicted box
        const float pcx = d0 * aw + acx;
        const float pcy = d1 * ah + acy;
        const float pw  = aw * __expf(fminf(d2, 4.0f));
        const float ph  = ah * __expf(fminf(d3, 4.0f));
        const float p1x = pcx - 0.5f * pw, p1y = pcy - 0.5f * ph;
        const float p2x = pcx + 0.5f * pw, p2y = pcy + 0.5f * ph;

        // decode target box
        const float tcx = tg.x * aw + acx;
        const float tcy = tg.y * ah + acy;
        const float tw  = aw * __expf(fminf(tg.z, 4.0f));
        const float th  = ah * __expf(fminf(tg.w, 4.0f));
        const float t1x = tcx - 0.5f * tw, t1y = tcy - 0.5f * th;
        const float t2x = tcx + 0.5f * tw, t2y = tcy + 0.5f * th;

        // aligned IoU
        const float ix    = fmaxf(0.f, fminf(p2x, t2x) - fmaxf(p1x, t1x));
        const float iy    = fmaxf(0.f, fminf(p2y, t2y) - fmaxf(p1y, t1y));
        const float inter = ix * iy;
        const float a1    = (p2x - p1x) * (p2y - p1y);
        const float a2    = (t2x - t1x) * (t2y - t1y);
        const float score = (inter / fmaxf(a1 + a2 - inter, 1e-9f)) * m;

        // smooth-L1 over positives
        float sl = smooth_l1(d0 - tg.x) + smooth_l1(d1 - tg.y)
                 + smooth_l1(d2 - tg.z) + smooth_l1(d3 - tg.w);
        locL = sl * m;
        posC = m;

        // ---- cls branch: QFL over 80 channels, coalesced across wave ----
        const float* clsptr = cls_p + (size_t)(b * A_ * C_ + a * C_) * stride + sp;
        const int tc = label - 1;      // -1 when background -> never matches
        float xmax = -3.402823466e38f;
        int   amax = 0;
        float cl   = 0.f;
        #pragma unroll 4
        for (int cc = 0; cc < C_; ++cc) {
            const float xv  = clsptr[(size_t)cc * stride];
            const float e   = __expf(-fabsf(xv));
            const float den = 1.0f + e;
            const float sig = (xv >= 0.f) ? (1.0f / den) : (e / den);
            const float spl = fmaxf(xv, 0.f) + __logf(den);   // softplus(xv)
            const float t   = (cc == tc) ? score : 0.f;
            const float dif = sig - t;
            cl += dif * dif * (spl - xv * t);
            if (xv > xmax) { xmax = xv; amax = cc; }          // first-max, like argmax
        }
        clsL = cl;

        const int pred = (xmax > 0.f) ? (amax + 1) : 0;       // sigmoid(x)>0.5 <=> x>0
        accC = (pred == label) ? 1.f : 0.f;
    }

    // all 256 threads reach this (EXEC all ones for WMMA)
    block_accum(clsL, locL, accC, posC, slots + (size_t)(slot_base + blockIdx.x) * 4);
}

__global__ __launch_bounds__(256)
void finalize_kernel(const float* __restrict__ slots, int nslots, float* __restrict__ out) {
    __shared__ float lds[256 * 4];
    float s0 = 0.f, s1 = 0.f, s2 = 0.f, s3 = 0.f;
    for (int i = threadIdx.x; i < nslots; i += 256) {   // fixed order per thread
        s0 += slots[i * 4 + 0];
        s1 += slots[i * 4 + 1];
        s2 += slots[i * 4 + 2];
        s3 += slots[i * 4 + 3];
    }
    lds[threadIdx.x * 4 + 0] = s0;
    lds[threadIdx.x * 4 + 1] = s1;
    lds[threadIdx.x * 4 + 2] = s2;
    lds[threadIdx.x * 4 + 3] = s3;
    __syncthreads();
    for (int st = 128; st > 0; st >>= 1) {              // fixed pairing -> deterministic
        if ((int)threadIdx.x < st) {
            #pragma unroll
            for (int q = 0; q < 4; ++q)
                lds[threadIdx.x * 4 + q] += lds[(threadIdx.x + st) * 4 + q];
        }
        __syncthreads();
    }
    if (threadIdx.x == 0) {
        const float pos_norm = fmaxf(1.0f, lds[3]);
        out[0] = lds[0] / pos_norm;                          // cls_loss
        out[1] = lds[1] / pos_norm;                          // loc_loss
        out[2] = lds[2] / (float)(BATCH * KTOT);             // accuracy
    }
}

extern "C" void kernel_launch(void* const* d_in, const int* in_sizes, int n_in,
                              void* d_out, int out_size, void* d_ws, size_t ws_size,
                              hipStream_t stream) {
    static const int HW[5][2] = {{100,100},{50,50},{25,25},{13,13},{7,7}};

    // dict order: cls_p3, loc_p3, cls_p4, loc_p4, ..., anchors, loc_target,
    //             cls_target, loc_mask
    const float* anchors  = (const float*)d_in[10];
    const float* loc_tgt  = (const float*)d_in[11];
    const int*   cls_tgt  = (const int*)d_in[12];
    const unsigned char* loc_mask = (const unsigned char*)d_in[13];

    float* slots = (float*)d_ws;

    int slot_base = 0;
    int k_base    = 0;
    for (int l = 0; l < 5; ++l) {
        const int h = HW[l][0], w = HW[l][1];
        const int n = BATCH * A_ * h * w;
        const int g = (n + 255) / 256;
        qfl_level_kernel<<<g, 256, 0, stream>>>(
            (const float*)d_in[2 * l], (const float*)d_in[2 * l + 1],
            anchors, loc_tgt, cls_tgt, loc_mask,
            slots, slot_base, h, w, k_base);
        slot_base += g;
        k_base    += A_ * h * w;
    }

    finalize_kernel<<<1, 256, 0, stream>>>(slots, slot_base, (float*)d_out);
}